// Attention_40278203302138
// MI455X (gfx1250) — compile-verified
//
#include <hip/hip_runtime.h>
#include <hip/hip_bf16.h>

// ---------------------------------------------------------------------------
// attn[b,s] = softmax_s( enc[b,s,:] . v[b,:] + c[b] )
//   v[b,:] = hidden[b,:] @ W      (W indexed [e,d]; v[d] = sum_e h[e]W[e,d])
//   c[b]   = hidden[b,:] . bias
// B=32, S=2048, D=512.  Memory-bound on the single 128MB read of enc.
// ---------------------------------------------------------------------------

#define BB 32
#define SS 2048
#define DD 512

typedef __attribute__((ext_vector_type(2))) float v2f;
typedef __attribute__((ext_vector_type(8))) float v8f;

// ---------------------------------------------------------------------------
// Kernel 1: v = h @ W via V_WMMA_F32_16X16X4_F32.
// Grid: 64 single-wave blocks. block = (m_tile in {0,1}) + 2*(n_tile in 0..31).
// A (16x4 f32) lane layout (ISA 7.12.2): lanes 0-15 hold M=lane, VGPR0=K0,
// VGPR1=K1; lanes 16-31 hold M=lane-16, VGPR0=K2, VGPR1=K3.
// B (4x16) analogous with K as the striped-row dim, N across lane halves.
// C/D (16x16 f32): VGPR j -> (M = j + 8*laneHalf, N = lane&15).
// ---------------------------------------------------------------------------
__global__ __launch_bounds__(32) void vgemm_wmma_kernel(
    const float* __restrict__ H,   // [32, 512] hidden
    const float* __restrict__ W,   // [512, 512], row index e, col index d
    float* __restrict__ V)         // [32, 512] out
{
  const int lane = threadIdx.x;        // 0..31, full wave => EXEC all ones
  const int mt   = blockIdx.x & 1;     // M tile
  const int nt   = blockIdx.x >> 1;    // N tile
  const int half = lane >> 4;          // 0 or 1
  const int l16  = lane & 15;

  const int m = mt * 16 + l16;         // A row handled by this lane
  const int n = nt * 16 + l16;         // B/C column handled by this lane

  v8f acc = {};
  for (int kk = 0; kk < DD; kk += 4) {
    const int ka = kk + 2 * half;      // {K0,K1} lanes 0-15, {K2,K3} lanes 16-31
    v2f a;
    a.x = H[m * DD + ka];
    a.y = H[m * DD + ka + 1];
    v2f b;
    b.x = W[(size_t)ka       * DD + n];
    b.y = W[(size_t)(ka + 1) * DD + n];
    // 8-arg form: (neg_a, A, neg_b, B, c_mod, C, reuse_a, reuse_b)
    acc = __builtin_amdgcn_wmma_f32_16x16x4_f32(
        false, a, false, b, (short)0, acc, false, false);
  }

  const int row0 = mt * 16 + half * 8;
  #pragma unroll
  for (int j = 0; j < 8; ++j)
    V[(size_t)(row0 + j) * DD + n] = acc[j];
}

// ---------------------------------------------------------------------------
// Kernel 2: c[b] = hidden[b,:] . bias   (one wave per batch)
// ---------------------------------------------------------------------------
__global__ __launch_bounds__(32) void hbias_kernel(
    const float* __restrict__ H, const float* __restrict__ bias,
    float* __restrict__ C)
{
  const int b    = blockIdx.x;
  const int lane = threadIdx.x;
  const float4* h4 = (const float4*)(H + (size_t)b * DD);
  const float4* g4 = (const float4*)bias;
  float acc = 0.f;
  #pragma unroll
  for (int i = 0; i < DD / (4 * 32); ++i) {  // 4 float4 per lane
    float4 h = h4[lane + 32 * i];
    float4 g = g4[lane + 32 * i];
    acc = fmaf(h.x, g.x, fmaf(h.y, g.y, fmaf(h.z, g.z, fmaf(h.w, g.w, acc))));
  }
  #pragma unroll
  for (int off = 16; off; off >>= 1) acc += __shfl_xor(acc, off, 32);
  if (lane == 0) C[b] = acc;
}

// ---------------------------------------------------------------------------
// Kernel 3 (the bandwidth kernel): scores[b,s] = enc[b,s,:] . v[b,:] + c[b]
// One wave per row of 512 floats; lane does 4x float4 (global_load_b128),
// coalesced across lanes. 8 waves / 256-thread block.
// ---------------------------------------------------------------------------
__global__ __launch_bounds__(256) void scores_kernel(
    const float* __restrict__ enc, const float* __restrict__ V,
    const float* __restrict__ C, float* __restrict__ scores)
{
  const int wave = threadIdx.x >> 5;
  const int lane = threadIdx.x & 31;
  const int row  = blockIdx.x * 8 + wave;       // 0 .. B*S-1
  const int b    = row >> 11;                   // / 2048

  const float4* e4 = (const float4*)(enc + (size_t)row * DD);
  const float4* v4 = (const float4*)(V + (size_t)b * DD);

  float acc = 0.f;
  #pragma unroll
  for (int i = 0; i < DD / (4 * 32); ++i) {
    float4 e = e4[lane + 32 * i];
    float4 w = v4[lane + 32 * i];
    acc = fmaf(e.x, w.x, fmaf(e.y, w.y, fmaf(e.z, w.z, fmaf(e.w, w.w, acc))));
  }
  #pragma unroll
  for (int off = 16; off; off >>= 1) acc += __shfl_xor(acc, off, 32);
  if (lane == 0) scores[row] = acc + C[b];
}

// ---------------------------------------------------------------------------
// Kernel 4: in-place softmax over S=2048 per batch. One block (256 thr) per b.
// ---------------------------------------------------------------------------
__global__ __launch_bounds__(256) void softmax_kernel(float* __restrict__ out)
{
  const int b = blockIdx.x;
  const int t = threadIdx.x;
  const int w = t >> 5, lane = t & 31;
  float* row = out + (size_t)b * SS;

  float vals[SS / 256];
  float m = -3.402823466e38f;
  #pragma unroll
  for (int j = 0; j < SS / 256; ++j) {
    vals[j] = row[t + 256 * j];
    m = fmaxf(m, vals[j]);
  }
  __shared__ float red[8];
  #pragma unroll
  for (int off = 16; off; off >>= 1) m = fmaxf(m, __shfl_xor(m, off, 32));
  if (lane == 0) red[w] = m;
  __syncthreads();
  if (w == 0) {
    float x = (lane < 8) ? red[lane] : -3.402823466e38f;
    #pragma unroll
    for (int off = 4; off; off >>= 1) x = fmaxf(x, __shfl_xor(x, off, 32));
    if (lane == 0) red[0] = x;
  }
  __syncthreads();
  m = red[0];
  __syncthreads();

  float s = 0.f;
  #pragma unroll
  for (int j = 0; j < SS / 256; ++j) {
    vals[j] = __expf(vals[j] - m);
    s += vals[j];
  }
  #pragma unroll
  for (int off = 16; off; off >>= 1) s += __shfl_xor(s, off, 32);
  if (lane == 0) red[w] = s;
  __syncthreads();
  if (w == 0) {
    float x = (lane < 8) ? red[lane] : 0.f;
    #pragma unroll
    for (int off = 4; off; off >>= 1) x += __shfl_xor(x, off, 32);
    if (lane == 0) red[0] = x;
  }
  __syncthreads();
  const float inv = 1.0f / red[0];
  #pragma unroll
  for (int j = 0; j < SS / 256; ++j) row[t + 256 * j] = vals[j] * inv;
}

// ---------------------------------------------------------------------------
extern "C" void kernel_launch(void* const* d_in, const int* in_sizes, int n_in,
                              void* d_out, int out_size, void* d_ws, size_t ws_size,
                              hipStream_t stream) {
  const float* hidden = (const float*)d_in[0];  // [32,1,512]
  const float* enc    = (const float*)d_in[1];  // [32,2048,512]
  const float* W      = (const float*)d_in[2];  // [512,512]
  const float* bias   = (const float*)d_in[3];  // [512]
  float* out = (float*)d_out;                   // [32,1,2048]

  float* v = (float*)d_ws;                      // 32*512 floats (64 KB)
  float* c = v + BB * DD;                       // 32 floats

  vgemm_wmma_kernel<<<dim3(64), dim3(32), 0, stream>>>(hidden, W, v);
  hbias_kernel<<<dim3(BB), dim3(32), 0, stream>>>(hidden, bias, c);
  scores_kernel<<<dim3((BB * SS) / 8), dim3(256), 0, stream>>>(enc, v, c, out);
  softmax_kernel<<<dim3(BB), dim3(256), 0, stream>>>(out);
}